// CL_31490700214617
// MI455X (gfx1250) — compile-verified
//
#include <hip/hip_runtime.h>
#include <hip/hip_bf16.h>

// Problem constants (match reference).
#define NROWS 4096
#define DIM   1024
#define NCLS  10
#define CPAD  16          // classes padded to 16 for the WMMA tile
#define COLS_PER_BLK 64
#define ROWS_PER_BLK 256
#define BLK 256

// Workspace layout (floats):
//   [0 , CPAD*DIM)        Sbar  (per-class sum of fm_s, padded rows zero)
//   [CPAD*DIM, 2*CPAD*DIM) Tbar
//   [2*CPAD*DIM, +CPAD)    cnt  (class counts as float)
//   next 2 floats          sc[0]=sum(s^2), sc[1]=sum(t^2)
#define WS_SBAR 0
#define WS_TBAR (CPAD * DIM)
#define WS_CNT  (2 * CPAD * DIM)
#define WS_SC   (WS_CNT + CPAD)

typedef __attribute__((ext_vector_type(2))) float v2f;
typedef __attribute__((ext_vector_type(8))) float v8f;

// ---------------------------------------------------------------------------
// Kernel 1: zero accumulators, build class histogram.
// ---------------------------------------------------------------------------
__global__ void cl_init_kernel(const int* __restrict__ targets, float* __restrict__ ws) {
    __shared__ int hist[CPAD];
    int t = threadIdx.x;
    if (t < CPAD) hist[t] = 0;
    __syncthreads();
    // Zero Sbar + Tbar (2 * 16 * 1024 floats).
    for (int i = t; i < 2 * CPAD * DIM; i += BLK) ws[i] = 0.0f;
    for (int i = t; i < NROWS; i += BLK) atomicAdd(&hist[targets[i]], 1);
    __syncthreads();
    if (t < CPAD) ws[WS_CNT + t] = (float)hist[t];
    if (t < 2)    ws[WS_SC + t] = 0.0f;
}

// ---------------------------------------------------------------------------
// Kernel 2: streaming pass. Block (bx,by) covers columns [bx*64,+64) and rows
// [by*256,+256). Thread t owns column (t&63) in row-group (t>>6); per-thread
// per-class accumulators live in LDS (no contention: each thread RMWs only
// its own 10 slots). One global atomic add per (class,col) partial at the end.
// ---------------------------------------------------------------------------
__global__ void cl_accum_kernel(const float* __restrict__ fm_s,
                                const float* __restrict__ fm_t,
                                const int*   __restrict__ targets,
                                float*       __restrict__ ws) {
    __shared__ float accS[BLK * NCLS];
    __shared__ float accT[BLK * NCLS];
    __shared__ int   tgt[ROWS_PER_BLK];
    __shared__ float red[BLK];

    const int t   = threadIdx.x;
    const int col = t & (COLS_PER_BLK - 1);
    const int g   = t >> 6;                      // row group 0..3
    const int colbase = blockIdx.x * COLS_PER_BLK;
    const int rowbase = blockIdx.y * ROWS_PER_BLK;

    #pragma unroll
    for (int k = 0; k < NCLS; ++k) { accS[t * NCLS + k] = 0.0f; accT[t * NCLS + k] = 0.0f; }
    tgt[t] = targets[rowbase + t];
    __syncthreads();

    const float* ps = fm_s + (size_t)rowbase * DIM + colbase + col;
    const float* pt = fm_t + (size_t)rowbase * DIM + colbase + col;

    float s2 = 0.0f, t2 = 0.0f;
    #pragma unroll 4
    for (int k = 0; k < ROWS_PER_BLK / 4; ++k) {
        const int il = g + 4 * k;                 // local row
        const float sv = ps[(size_t)il * DIM];
        const float tv = pt[(size_t)il * DIM];
        const int c = tgt[il];
        accS[t * NCLS + c] += sv;
        accT[t * NCLS + c] += tv;
        s2 += sv * sv;
        t2 += tv * tv;
    }

    // Block-reduce s2, t2 -> single atomic each.
    red[t] = s2; __syncthreads();
    for (int off = BLK / 2; off > 0; off >>= 1) { if (t < off) red[t] += red[t + off]; __syncthreads(); }
    if (t == 0) atomicAdd(&ws[WS_SC + 0], red[0]);
    __syncthreads();
    red[t] = t2; __syncthreads();
    for (int off = BLK / 2; off > 0; off >>= 1) { if (t < off) red[t] += red[t + off]; __syncthreads(); }
    if (t == 0) atomicAdd(&ws[WS_SC + 1], red[0]);
    __syncthreads();

    // Fold the 4 row-groups and commit (class, col) partials.
    for (int p = t; p < NCLS * COLS_PER_BLK; p += BLK) {
        const int cls = p / COLS_PER_BLK;
        const int c2  = p % COLS_PER_BLK;
        float vs = 0.0f, vt = 0.0f;
        #pragma unroll
        for (int gg = 0; gg < 4; ++gg) {
            const int tt = gg * COLS_PER_BLK + c2;
            vs += accS[tt * NCLS + cls];
            vt += accT[tt * NCLS + cls];
        }
        atomicAdd(&ws[WS_SBAR + cls * DIM + colbase + c2], vs);
        atomicAdd(&ws[WS_TBAR + cls * DIM + colbase + c2], vt);
    }
}

// ---------------------------------------------------------------------------
// Kernel 3: single wave32. Gram tile diag via V_WMMA_F32_16X16X4_F32:
//   C[16x16] = Sbar[16x1024] * Tbar^T[1024x16], take diag(c)/cnt[c].
// A 16x4 layout (ISA table): lane = M (+16 for K-half), VGPR0/1 = consecutive K.
// B 4x16: VGPR v holds row K = v + (lane>=16 ? 2 : 0), N = lane%16.
// Both operand fetch patterns collapse to one aligned b64 load per matrix.
// ---------------------------------------------------------------------------
__global__ void cl_finish_kernel(const float* __restrict__ ws, float* __restrict__ out) {
    __shared__ float tile[16][16];
    const float* Sbar = ws + WS_SBAR;
    const float* Tbar = ws + WS_TBAR;
    const float* cnt  = ws + WS_CNT;
    const float* sc   = ws + WS_SC;

    const int lane  = threadIdx.x;          // 0..31, EXEC all ones
    const int m     = lane & 15;            // A row / B column index
    const int khalf = (lane >> 4) << 1;     // 0 for lanes 0-15, 2 for 16-31

    v8f c = {};
    for (int kb = 0; kb < DIM; kb += 4) {
        const v2f a = *(const v2f*)(Sbar + (size_t)m * DIM + kb + khalf);
        const v2f b = *(const v2f*)(Tbar + (size_t)m * DIM + kb + khalf);
        c = __builtin_amdgcn_wmma_f32_16x16x4_f32(
                /*neg_a=*/false, a, /*neg_b=*/false, b,
                /*c_mod=*/(short)0, c, /*reuse_a=*/false, /*reuse_b=*/false);
    }

    // C/D layout: VGPR v -> M = v (+8 for lanes 16-31), N = lane%16.
    const int n     = lane & 15;
    const int mbase = (lane >> 4) << 3;
    #pragma unroll
    for (int v = 0; v < 8; ++v) tile[mbase + v][n] = c[v];
    __syncthreads();

    if (lane == 0) {
        float acc = 0.0f;
        for (int cc = 0; cc < NCLS; ++cc) {
            const float ct = cnt[cc];
            if (ct > 0.0f) acc += tile[cc][cc] / ct;
        }
        out[0] = (sc[0] + sc[1] - 2.0f * acc) / (float)NROWS;
    }
}

// ---------------------------------------------------------------------------
extern "C" void kernel_launch(void* const* d_in, const int* in_sizes, int n_in,
                              void* d_out, int out_size, void* d_ws, size_t ws_size,
                              hipStream_t stream) {
    (void)in_sizes; (void)n_in; (void)out_size; (void)ws_size;
    const float* fm_s    = (const float*)d_in[0];
    const float* fm_t    = (const float*)d_in[1];
    const int*   targets = (const int*)d_in[2];
    // d_in[3] (fusion_true) is the False branch in the reference -> unused.
    float* out = (float*)d_out;
    float* ws  = (float*)d_ws;

    cl_init_kernel<<<1, BLK, 0, stream>>>(targets, ws);
    dim3 grid(DIM / COLS_PER_BLK, NROWS / ROWS_PER_BLK);   // 16 x 16 blocks
    cl_accum_kernel<<<grid, BLK, 0, stream>>>(fm_s, fm_t, targets, ws);
    cl_finish_kernel<<<1, 32, 0, stream>>>(ws, out);
}